// DescriptorMatcherWithSteerer_45603962749386
// MI455X (gfx1250) — compile-verified
//
#include <hip/hip_runtime.h>

typedef __attribute__((ext_vector_type(2))) float v2f;
typedef __attribute__((ext_vector_type(8))) float v8f;

#define DDIM  128
#define NVER  4
#define TILE  16
#define WAVES 8
#define NTHR  (WAVES * 32)

// ---------------- steering: out[i][j] = sum_k in[i][k] * G[j][k]  (x @ G^T) ----------------
__global__ void steer_kernel(const float* __restrict__ in, const float* __restrict__ g,
                             float* __restrict__ out, int rows) {
    int tid = blockIdx.x * blockDim.x + threadIdx.x;
    if (tid >= rows * DDIM) return;
    int i = tid >> 7;          // / 128
    int j = tid & (DDIM - 1);  // % 128
    const float* xr = in + (size_t)i * DDIM;
    const float* gr = g + (size_t)j * DDIM;
    float acc = 0.f;
#pragma unroll 8
    for (int k = 0; k < DDIM; ++k) acc = fmaf(xr[k], gr[k], acc);
    out[tid] = acc;
}

// ---------------- squared norms for all 4 steered versions + desc2 ----------------
__global__ void norms_kernel(const float* __restrict__ steer, const float* __restrict__ desc2,
                             float* __restrict__ n1, float* __restrict__ n2, int B1, int B2) {
    int tid = blockIdx.x * blockDim.x + threadIdx.x;
    int total = NVER * B1 + B2;
    if (tid >= total) return;
    const float* row = (tid < NVER * B1) ? (steer + (size_t)tid * DDIM)
                                         : (desc2 + (size_t)(tid - NVER * B1) * DDIM);
    const float4* r4 = (const float4*)row;
    float acc = 0.f;
#pragma unroll
    for (int k = 0; k < DDIM / 4; ++k) {
        float4 v = r4[k];
        acc = fmaf(v.x, v.x, acc); acc = fmaf(v.y, v.y, acc);
        acc = fmaf(v.z, v.z, acc); acc = fmaf(v.w, v.w, acc);
    }
    if (tid < NVER * B1) n1[tid] = acc; else n2[tid - NVER * B1] = acc;
}

// ---------------- fused distance / min-over-steer / row argmin ----------------
__global__ __launch_bounds__(NTHR)
void match_kernel(const float* __restrict__ steer, const float* __restrict__ desc2,
                  const float* __restrict__ n1, const float* __restrict__ n2,
                  float* __restrict__ out, int B1, int B2) {
    __shared__ float ldsA[NVER][TILE][DDIM];   // 32 KB: 4 steered 16x128 A tiles
    __shared__ float ldsN1[NVER][TILE];
    __shared__ float redMin[WAVES][8][32];
    __shared__ int   redArg[WAVES][8][32];

    const int i0   = blockIdx.x * TILE;
    const int t    = threadIdx.x;
    const int lane = t & 31;
    const int wave = t >> 5;
    const int n    = lane & 15;   // column within 16x16 C tile (and A row m)
    const int kh   = lane >> 4;   // K-pair half select per ISA WMMA layout

    // stage A tiles (all 4 versions) into LDS, float4 cooperative loads
    for (int idx = t; idx < NVER * TILE * (DDIM / 4); idx += NTHR) {
        int v   = idx / (TILE * (DDIM / 4));
        int rem = idx - v * (TILE * (DDIM / 4));
        int r   = rem / (DDIM / 4);
        int k4  = rem - r * (DDIM / 4);
        int gi  = i0 + r; if (gi >= B1) gi = B1 - 1;
        ((float4*)&ldsA[v][r][0])[k4] =
            ((const float4*)(steer + ((size_t)v * B1 + gi) * DDIM))[k4];
    }
    for (int idx = t; idx < NVER * TILE; idx += NTHR) {
        int v = idx >> 4, r = idx & 15;
        int gi = i0 + r; if (gi >= B1) gi = B1 - 1;
        ldsN1[v][r] = n1[(size_t)v * B1 + gi];
    }
    __syncthreads();

    float rowmin[8]; int rowarg[8];
#pragma unroll
    for (int r = 0; r < 8; ++r) { rowmin[r] = __builtin_inff(); rowarg[r] = 0; }

    // per-lane A fragment pointers (row m = n, K-offset 2*kh), stride 4 floats per K-step
    const v2f* ap0 = (const v2f*)&ldsA[0][n][2 * kh];
    const v2f* ap1 = (const v2f*)&ldsA[1][n][2 * kh];
    const v2f* ap2 = (const v2f*)&ldsA[2][n][2 * kh];
    const v2f* ap3 = (const v2f*)&ldsA[3][n][2 * kh];

    const int jtiles = (B2 + TILE - 1) / TILE;
    for (int jt = wave; jt < jtiles; jt += WAVES) {
        const int j0 = jt * TILE;
        int  gj     = j0 + n;
        bool jvalid = (gj < B2);
        int  gjc    = jvalid ? gj : (B2 - 1);
        const float* brow = desc2 + (size_t)gjc * DDIM;
        float n2j = jvalid ? n2[gjc] : __builtin_inff();

        // prefetch next j-tile's B row for this lane into cache hierarchy
        int jn = j0 + WAVES * TILE + n;
        if (jn < B2) __builtin_prefetch(desc2 + (size_t)jn * DDIM, 0, 1);

        const v2f* bp = (const v2f*)(brow + 2 * kh);

        v8f acc0 = {}, acc1 = {}, acc2 = {}, acc3 = {};
#pragma unroll 4
        for (int ks = 0; ks < DDIM / 4; ++ks) {
            v2f b  = bp[2 * ks];       // B fragment: 4x16 tile, rows of desc2 as columns
            v2f a0 = ap0[2 * ks];
            v2f a1 = ap1[2 * ks];
            v2f a2 = ap2[2 * ks];
            v2f a3 = ap3[2 * ks];
            acc0 = __builtin_amdgcn_wmma_f32_16x16x4_f32(false, a0, false, b, (short)0, acc0, false, false);
            acc1 = __builtin_amdgcn_wmma_f32_16x16x4_f32(false, a1, false, b, (short)0, acc1, false, false);
            acc2 = __builtin_amdgcn_wmma_f32_16x16x4_f32(false, a2, false, b, (short)0, acc2, false, false);
            acc3 = __builtin_amdgcn_wmma_f32_16x16x4_f32(false, a3, false, b, (short)0, acc3, false, false);
        }

        // C/D layout: VGPR r -> row (r + 8*kh), column n; fold versions + running argmin
#pragma unroll
        for (int r = 0; r < 8; ++r) {
            int row = r + 8 * kh;
            float d0 = ldsN1[0][row] + n2j - 2.f * acc0[r];
            float d1 = ldsN1[1][row] + n2j - 2.f * acc1[r];
            float d2 = ldsN1[2][row] + n2j - 2.f * acc2[r];
            float d3 = ldsN1[3][row] + n2j - 2.f * acc3[r];
            float dmin = fmaxf(fminf(fminf(d0, d1), fminf(d2, d3)), 0.f);
            if (dmin < rowmin[r]) { rowmin[r] = dmin; rowarg[r] = gj; }  // strict < keeps lowest j
        }
    }

#pragma unroll
    for (int r = 0; r < 8; ++r) {
        redMin[wave][r][lane] = rowmin[r];
        redArg[wave][r][lane] = rowarg[r];
    }
    __syncthreads();

    // 16 threads finalize the 16 rows of this i-tile
    if (t < TILE) {
        int row = t;
        int r   = row & 7;
        int lo  = (row < 8) ? 0 : 16;   // lanes 0-15 hold M=r, lanes 16-31 hold M=r+8
        float best = __builtin_inff(); int barg = 0x7fffffff;
        for (int w = 0; w < WAVES; ++w)
            for (int l = lo; l < lo + 16; ++l) {
                float v = redMin[w][r][l]; int a = redArg[w][r][l];
                if (v < best || (v == best && a < barg)) { best = v; barg = a; }
            }
        int gi = i0 + row;
        if (gi < B1) {
            out[gi] = sqrtf(best);                    // match_dists (B1,1)
            out[B1 + 2 * gi + 0] = (float)gi;         // matches_idxs[:,0]
            out[B1 + 2 * gi + 1] = (float)barg;       // matches_idxs[:,1]
        }
    }
}

extern "C" void kernel_launch(void* const* d_in, const int* in_sizes, int n_in,
                              void* d_out, int out_size, void* d_ws, size_t ws_size,
                              hipStream_t stream) {
    const float* desc1 = (const float*)d_in[0];
    const float* desc2 = (const float*)d_in[1];
    const float* gen   = (const float*)d_in[2];
    // d_in[3] = steerer_order, fixed at 4 by the reference (NVER compiled in)

    const int B1 = in_sizes[0] / DDIM;
    const int B2 = in_sizes[1] / DDIM;

    float* steer = (float*)d_ws;                              // NVER * B1 * D floats
    float* n1    = steer + (size_t)NVER * B1 * DDIM;          // NVER * B1
    float* n2v   = n1 + (size_t)NVER * B1;                    // B2
    float* out   = (float*)d_out;

    // version 0 = identity copy of desc1
    hipMemcpyAsync(steer, desc1, (size_t)B1 * DDIM * sizeof(float),
                   hipMemcpyDeviceToDevice, stream);

    const int thr = 256;
    const int steerBlocks = (B1 * DDIM + thr - 1) / thr;
    for (int v = 0; v < NVER - 1; ++v)
        steer_kernel<<<steerBlocks, thr, 0, stream>>>(
            steer + (size_t)v * B1 * DDIM, gen, steer + (size_t)(v + 1) * B1 * DDIM, B1);

    const int totalRows = NVER * B1 + B2;
    norms_kernel<<<(totalRows + thr - 1) / thr, thr, 0, stream>>>(steer, desc2, n1, n2v, B1, B2);

    const int iblocks = (B1 + TILE - 1) / TILE;
    match_kernel<<<iblocks, NTHR, 0, stream>>>(steer, desc2, n1, n2v, out, B1, B2);
}